// CondGCN_72696616452545
// MI455X (gfx1250) — compile-verified
//
#include <hip/hip_runtime.h>

// CondGCN on MI455X (gfx1250, wave32).
// Rewrite: relu(Linear(gather(x))) == gather(relu(Linear(x))) (row-wise map
// commutes with gather) -> per-node transforms once via V_WMMA_F32_16X16X4_F32
// (exact fp32 matrix path), edges become gather/scatter-add of 64-float rows
// (all operands L2-resident in the 192MB L2; scatter = global_atomic_add_f32).

typedef __attribute__((ext_vector_type(2))) float v2f;
typedef __attribute__((ext_vector_type(8))) float v8f;

#define HID 64

// out[N,64] = (RELU?) (feat[N,FIN] @ W[64,FIN]^T + b[64])
// One wave -> 16 rows x 64 cols (4 accumulators of 16x16 f32).
// FIN is a compile-time constant -> fully unrolled k-loop, constant-offset
// clauses of global_load_b64 feeding back-to-back v_wmma.
template <int FIN, bool RELU>
__global__ __launch_bounds__(128) void linrelu_wmma_kernel(
    const float* __restrict__ feat, const float* __restrict__ W,
    const float* __restrict__ bias, float* __restrict__ out, int N)
{
    const int lane = threadIdx.x & 31;
    const int wave = threadIdx.x >> 5;
    const int row0 = (blockIdx.x * 4 + wave) * 16;
    if (row0 >= N) return;               // wave-uniform: EXEC stays all-1s below

    const int half = lane >> 4;          // 0: lanes 0-15, 1: lanes 16-31
    const int l16  = lane & 15;
    const int kb   = half * 2;           // this lane's K offset inside a 4-step

    // A fragment source row (clamped for the tail tile; stores are masked)
    int arow = row0 + l16;
    if (arow >= N) arow = N - 1;
    const float* aptr = feat + (size_t)arow * FIN + kb;

    // Independent B base pointers for the four 16-col tiles:
    // B element (K=kk+kb+j, N=16t+l16) = W[(16t+l16)*FIN + kk+kb+j]
    const float* wp0 = W + (size_t)(l16 +  0) * FIN + kb;
    const float* wp1 = W + (size_t)(l16 + 16) * FIN + kb;
    const float* wp2 = W + (size_t)(l16 + 32) * FIN + kb;
    const float* wp3 = W + (size_t)(l16 + 48) * FIN + kb;

    v8f acc0 = {}, acc1 = {}, acc2 = {}, acc3 = {};

#pragma unroll
    for (int kk = 0; kk < FIN; kk += 4) {
        // 5 independent b64 loads (clause-able), then 4 independent WMMAs
        const v2f a  = *(const v2f*)(aptr + kk);
        const v2f b0 = *(const v2f*)(wp0 + kk);
        const v2f b1 = *(const v2f*)(wp1 + kk);
        const v2f b2 = *(const v2f*)(wp2 + kk);
        const v2f b3 = *(const v2f*)(wp3 + kk);
        acc0 = __builtin_amdgcn_wmma_f32_16x16x4_f32(false, a, false, b0, (short)0, acc0, false, false);
        acc1 = __builtin_amdgcn_wmma_f32_16x16x4_f32(false, a, false, b1, (short)0, acc1, false, false);
        acc2 = __builtin_amdgcn_wmma_f32_16x16x4_f32(false, a, false, b2, (short)0, acc2, false, false);
        acc3 = __builtin_amdgcn_wmma_f32_16x16x4_f32(false, a, false, b3, (short)0, acc3, false, false);
    }

    // C/D layout: VGPR i -> row = row0 + i + 8*half, col = 16t + l16
    v8f accs[4] = {acc0, acc1, acc2, acc3};
#pragma unroll
    for (int t = 0; t < 4; ++t) {
        const int col = 16 * t + l16;
        const float bv = bias[col];
#pragma unroll
        for (int i = 0; i < 8; ++i) {
            const int row = row0 + i + 8 * half;
            if (row < N) {
                float v = accs[t][i] + bv;
                if (RELU) v = fmaxf(v, 0.0f);
                out[(size_t)row * HID + col] = v;
            }
        }
    }
}

// agg[dst] += h[src] over E edges. One wave covers one edge's 64 floats
// (2 per lane) -> coalesced b64 gather, broadcast index loads, hw f32 atomics.
__global__ __launch_bounds__(256) void scatter_add_kernel(
    const float* __restrict__ h, const int* __restrict__ src,
    const int* __restrict__ dst, float* __restrict__ agg, int E)
{
    const long long total  = (long long)E * 32;
    const long long stride = (long long)gridDim.x * blockDim.x;
    for (long long i = (long long)blockIdx.x * blockDim.x + threadIdx.x;
         i < total; i += stride) {
        const int e = (int)(i >> 5);
        const int p = (int)(i & 31);
        const int s = src[e];
        const int d = dst[e];
        const float2 v = *(const float2*)(h + (size_t)s * HID + p * 2);
        float* o = agg + (size_t)d * HID + p * 2;
        __hip_atomic_fetch_add(o,     v.x, __ATOMIC_RELAXED, __HIP_MEMORY_SCOPE_AGENT);
        __hip_atomic_fetch_add(o + 1, v.y, __ATOMIC_RELAXED, __HIP_MEMORY_SCOPE_AGENT);
    }
}

extern "C" void kernel_launch(void* const* d_in, const int* in_sizes, int n_in,
                              void* d_out, int out_size, void* d_ws, size_t ws_size,
                              hipStream_t stream) {
    const float* x = (const float*)d_in[0];
    const float* c = (const float*)d_in[1];
    const float* r = (const float*)d_in[2];
    const int* exx_s = (const int*)d_in[3];
    const int* exx_d = (const int*)d_in[4];
    const int* ecx_s = (const int*)d_in[5];
    const int* ecx_d = (const int*)d_in[6];
    const int* erx_s = (const int*)d_in[7];
    const int* erx_d = (const int*)d_in[8];
    const float* W_x    = (const float*)d_in[9];  const float* b_x    = (const float*)d_in[10];
    const float* W_c    = (const float*)d_in[11]; const float* b_c    = (const float*)d_in[12];
    const float* W_r    = (const float*)d_in[13]; const float* b_r    = (const float*)d_in[14];
    const float* W_xx   = (const float*)d_in[15]; const float* b_xx   = (const float*)d_in[16];
    const float* W_cx   = (const float*)d_in[17]; const float* b_cx   = (const float*)d_in[18];
    const float* W_rx   = (const float*)d_in[19]; const float* b_rx   = (const float*)d_in[20];
    const float* W_pool = (const float*)d_in[21]; const float* b_pool = (const float*)d_in[22];

    const int FX = 64, FC = 32, FR = 48;
    const int NX = in_sizes[0] / FX;
    const int NC = in_sizes[1] / FC;
    const int NR = in_sizes[2] / FR;
    const int EXX = in_sizes[3];
    const int ECX = in_sizes[5];
    const int ERX = in_sizes[7];

    // workspace layout (fp32): agg[NX*64] | h_xx[NX*64] | h_cx[NC*64] | h_rx[NR*64]
    float* agg  = (float*)d_ws;
    float* h_xx = agg  + (size_t)NX * HID;
    float* h_cx = h_xx + (size_t)NX * HID;
    float* h_rx = h_cx + (size_t)NC * HID;

    float* x_out = (float*)d_out;
    float* c_out = x_out + (size_t)NX * HID;
    float* r_out = c_out + (size_t)NC * HID;

    const dim3 blk(128);
    const dim3 gx((unsigned)((NX + 63) / 64));
    const dim3 gc((unsigned)((NC + 63) / 64));
    const dim3 gr((unsigned)((NR + 63) / 64));

    // 1) per-node transforms (self_x goes straight into agg -> initializes it)
    linrelu_wmma_kernel<64, true><<<gx, blk, 0, stream>>>(x, W_x,  b_x,  agg,  NX);
    linrelu_wmma_kernel<64, true><<<gx, blk, 0, stream>>>(x, W_xx, b_xx, h_xx, NX);
    linrelu_wmma_kernel<32, true><<<gc, blk, 0, stream>>>(c, W_cx, b_cx, h_cx, NC);
    linrelu_wmma_kernel<48, true><<<gr, blk, 0, stream>>>(r, W_rx, b_rx, h_rx, NR);

    // 2) scatter-add messages into agg (L2-resident f32 atomics)
    auto sgrid = [](long long E) {
        long long b = (E * 32 + 255) / 256;
        if (b > 262144) b = 262144;
        return dim3((unsigned)b);
    };
    scatter_add_kernel<<<sgrid(EXX), dim3(256), 0, stream>>>(h_xx, exx_s, exx_d, agg, EXX);
    scatter_add_kernel<<<sgrid(ECX), dim3(256), 0, stream>>>(h_cx, ecx_s, ecx_d, agg, ECX);
    scatter_add_kernel<<<sgrid(ERX), dim3(256), 0, stream>>>(h_rx, erx_s, erx_d, agg, ERX);

    // 3) outputs: pool GEMM (no relu) + the two independent relu-linears
    linrelu_wmma_kernel<64, false><<<gx, blk, 0, stream>>>(agg, W_pool, b_pool, x_out, NX);
    linrelu_wmma_kernel<32, true ><<<gc, blk, 0, stream>>>(c,   W_c,    b_c,    c_out, NC);
    linrelu_wmma_kernel<48, true ><<<gr, blk, 0, stream>>>(r,   W_r,    b_r,    r_out, NR);
}